// AutoregU_23244363005995
// MI455X (gfx1250) — compile-verified
//
#include <hip/hip_runtime.h>
#include <math.h>

// Problem constants (match reference)
#define T_STEPS 6
#define NN 20000
#define EE 320000
#define INF 11
#define HID 64
#define OUTF 3

typedef float v2f __attribute__((ext_vector_type(2)));
typedef float v8f __attribute__((ext_vector_type(8)));

__device__ __forceinline__ v8f wmma_f32_k4(v2f a, v2f b, v8f c) {
    // V_WMMA_F32_16X16X4_F32 : D = A(16x4) * B(4x16) + C(16x16), all fp32
    return __builtin_amdgcn_wmma_f32_16x16x4_f32(
        /*neg_a=*/false, a, /*neg_b=*/false, b,
        /*c_mod=*/(short)0, c, /*reuse_a=*/false, /*reuse_b=*/false);
}

#define LDS_PITCH 65  // 64 + 1 pad: row-major 16-lane reads hit distinct LDS banks

// ---------------- utility kernels ----------------

__global__ void zero_kernel(float* __restrict__ p, int n) {
    int i = blockIdx.x * blockDim.x + threadIdx.x;
    if (i < n) p[i] = 0.0f;
}

__global__ void deg_kernel(const int* __restrict__ src, float* __restrict__ deg, int E) {
    int e = blockIdx.x * blockDim.x + threadIdx.x;
    if (e < E) atomicAdd(&deg[src[e]], 1.0f);
}

__global__ void dis_kernel(const float* __restrict__ deg, float* __restrict__ dis, int N) {
    int n = blockIdx.x * blockDim.x + threadIdx.x;
    if (n < N) {
        float d = deg[n];
        dis[n] = (d > 0.0f) ? rsqrtf(fmaxf(d, 1.0f)) : 0.0f;
    }
}

__global__ void ew_kernel(const int* __restrict__ src, const int* __restrict__ dst,
                          const float* __restrict__ dis, float* __restrict__ ew, int E) {
    int e = blockIdx.x * blockDim.x + threadIdx.x;
    if (e < E) ew[e] = -dis[src[e]] * dis[dst[e]];
}

// Build fused weight matrices:
//  Wzr  [128 x 128]: rows 0-63 = Wh[g][0], rows 64-127 = Wh[g][1]; cols 0-63 g=z, 64-127 g=r
//  Whh  [128 x  64]: rows 0-63 = Wh[2][0], rows 64-127 = Wh[2][1]
//  Wxall[ 22 x 192]: rows 0-10 = Wx[g][0], rows 11-21 = Wx[g][1]; col block g*64
__global__ void prep_w_kernel(const float* __restrict__ Wx, const float* __restrict__ Wh,
                              float* __restrict__ Wzr, float* __restrict__ Whh,
                              float* __restrict__ Wxall) {
    int i = blockIdx.x * blockDim.x + threadIdx.x;
    const int SZR = 128 * 128, SHH = 128 * 64, SXA = 22 * 192;
    if (i < SZR) {
        int k = i >> 7, c = i & 127;
        int g = c >> 6, cc = c & 63;
        int kk = k >> 6, kr = k & 63;
        Wzr[i] = Wh[(((g * 2 + kk) * 64) + kr) * 64 + cc];
    } else if (i < SZR + SHH) {
        int j = i - SZR;
        int k = j >> 6, c = j & 63;
        int kk = k >> 6, kr = k & 63;
        Whh[j] = Wh[(((2 * 2 + kk) * 64) + kr) * 64 + c];
    } else if (i < SZR + SHH + SXA) {
        int j = i - SZR - SHH;
        int k = j / 192, c = j - (j / 192) * 192;
        int g = c / 64, cc = c - g * 64;
        int kk = (k >= 11) ? 1 : 0, kr = k - 11 * kk;
        Wxall[j] = Wx[(((g * 2 + kk) * 11) + kr) * 64 + cc];
    }
}

// x_in construction: t==0 -> copy; t>0 -> cols 3:6 = u_prev, cols 8:11 = (u_prev - x_prev[3:6])/dt
__global__ void build_x_kernel(const float* __restrict__ X_seq,
                               const float* __restrict__ u_prev,
                               float* __restrict__ x_in, int t, int N) {
    int n = blockIdx.x * blockDim.x + threadIdx.x;
    if (n >= N) return;
    const float* xt = X_seq + ((size_t)t * N + n) * INF;
    float row[INF];
#pragma unroll
    for (int j = 0; j < INF; ++j) row[j] = xt[j];
    if (t > 0) {
        const float* xp = X_seq + ((size_t)(t - 1) * N + n) * INF;
        float dt = row[6] - xp[6];
#pragma unroll
        for (int j = 0; j < 3; ++j) {
            float u = u_prev[n * 3 + j];
            row[3 + j] = u;
            row[8 + j] = (u - xp[3 + j]) / dt;
        }
    }
#pragma unroll
    for (int j = 0; j < INF; ++j) x_in[n * INF + j] = row[j];
}

// Scatter SpMM: out[dst] += ew * x[src], F features per node
__global__ void spmm_kernel(const float* __restrict__ x, const int* __restrict__ src,
                            const int* __restrict__ dst, const float* __restrict__ ew,
                            float* __restrict__ out, int E, int F) {
    int idx = blockIdx.x * blockDim.x + threadIdx.x;
    if (idx >= E * F) return;
    int e = idx / F, f = idx - e * F;
    atomicAdd(&out[dst[e] * F + f], ew[e] * x[src[e] * F + f]);
}

// Gx[n, 0:192] = bx + x_in @ Wxall[0:11] + Tx @ Wxall[11:22]  (K=22, VALU path)
__global__ void gx_kernel(const float* __restrict__ x_in, const float* __restrict__ Tx,
                          const float* __restrict__ Wxall, const float* __restrict__ bx,
                          float* __restrict__ Gx, int N) {
    int idx = blockIdx.x * blockDim.x + threadIdx.x;
    if (idx >= N * 192) return;
    int n = idx / 192, c = idx - n * 192;
    float acc = bx[c];
#pragma unroll
    for (int k = 0; k < 11; ++k) {
        acc = fmaf(x_in[n * INF + k], Wxall[k * 192 + c], acc);
        acc = fmaf(Tx[n * INF + k], Wxall[(11 + k) * 192 + c], acc);
    }
    Gx[idx] = acc;
}

// ---------------- WMMA gate kernels ----------------
// ZR: pre = Gx[:,0:128] + bh[0:128] + [h|Th] @ Wzr ; Z = sigmoid(pre[:,0:64]);
// R = sigmoid(pre[:,64:128]); hr = h * R.
// Block = 256 threads (8 waves), 16 rows; wave w computes column tile w of the 128-wide output.
// A slabs (h, Th rows) staged once per block into LDS via coalesced loads; WMMA A-frags come
// from conflict-free ds reads (row pitch 65).
__global__ __launch_bounds__(256) void zr_kernel(
    const float* __restrict__ h, const float* __restrict__ Th,
    const float* __restrict__ Wzr, const float* __restrict__ Gx,
    const float* __restrict__ bzr, float* __restrict__ Z, float* __restrict__ hr, int N) {
    __shared__ float hs[16][LDS_PITCH];
    __shared__ float ts[16][LDS_PITCH];

    int lane = threadIdx.x & 31;
    int wave = threadIdx.x >> 5;           // 0..7 -> column tile
    int row0 = blockIdx.x * 16;

    // Cooperative coalesced stage: 256 threads x 4 elems = 16x64 slab, for h and Th
#pragma unroll
    for (int j = 0; j < 4; ++j) {
        int idx = threadIdx.x + j * 256;   // 0..1023
        int r = idx >> 6, c = idx & 63;
        hs[r][c] = h[(size_t)(row0 + r) * HID + c];
        ts[r][c] = Th[(size_t)(row0 + r) * HID + c];
    }
    __syncthreads();

    int nn = lane & 15;
    int kg = (lane >> 4) * 2;              // A/B fragment K sub-offset
    int mh = (lane >> 4) * 8;              // C/D fragment M sub-offset
    int col = wave * 16 + nn;              // 0..127

    v8f acc;
#pragma unroll
    for (int v = 0; v < 8; ++v)
        acc[v] = Gx[(size_t)(row0 + v + mh) * 192 + col] + bzr[col];

#pragma unroll
    for (int k0 = 0; k0 < 64; k0 += 4) {   // h @ Wzr[0:64]
        v2f a, b;
        a.x = hs[nn][k0 + kg];
        a.y = hs[nn][k0 + kg + 1];
        b.x = Wzr[(k0 + kg) * 128 + col];
        b.y = Wzr[(k0 + kg + 1) * 128 + col];
        acc = wmma_f32_k4(a, b, acc);
    }
#pragma unroll
    for (int k0 = 0; k0 < 64; k0 += 4) {   // Th @ Wzr[64:128]
        v2f a, b;
        a.x = ts[nn][k0 + kg];
        a.y = ts[nn][k0 + kg + 1];
        b.x = Wzr[(64 + k0 + kg) * 128 + col];
        b.y = Wzr[(64 + k0 + kg + 1) * 128 + col];
        acc = wmma_f32_k4(a, b, acc);
    }

#pragma unroll
    for (int v = 0; v < 8; ++v) {
        int row = row0 + v + mh;
        float s = 1.0f / (1.0f + expf(-acc[v]));
        if (col < 64) {
            Z[(size_t)row * HID + col] = s;
        } else {
            int c = col - 64;
            hr[(size_t)row * HID + c] = hs[v + mh][c] * s;
        }
    }
}

// Hnew: pre = Gx[:,128:192] + bh2 + [hr|Thr] @ Whh ; Ht = tanh(pre);
// h = Z*h + (1-Z)*Ht  (in-place, each element touched by exactly one lane)
// Block = 128 threads (4 waves), 16 rows; A slabs staged in LDS.
__global__ __launch_bounds__(128) void hnew_kernel(
    const float* __restrict__ hr, const float* __restrict__ Thr,
    const float* __restrict__ Whh, const float* __restrict__ Gx,
    const float* __restrict__ bh2, const float* __restrict__ Z,
    float* __restrict__ h, int N) {
    __shared__ float rs[16][LDS_PITCH];
    __shared__ float ts[16][LDS_PITCH];

    int lane = threadIdx.x & 31;
    int wave = threadIdx.x >> 5;           // 0..3
    int row0 = blockIdx.x * 16;

#pragma unroll
    for (int j = 0; j < 8; ++j) {
        int idx = threadIdx.x + j * 128;   // 0..1023
        int r = idx >> 6, c = idx & 63;
        rs[r][c] = hr[(size_t)(row0 + r) * HID + c];
        ts[r][c] = Thr[(size_t)(row0 + r) * HID + c];
    }
    __syncthreads();

    int nn = lane & 15;
    int kg = (lane >> 4) * 2;
    int mh = (lane >> 4) * 8;
    int col = wave * 16 + nn;              // 0..63

    v8f acc;
#pragma unroll
    for (int v = 0; v < 8; ++v)
        acc[v] = Gx[(size_t)(row0 + v + mh) * 192 + 128 + col] + bh2[col];

#pragma unroll
    for (int k0 = 0; k0 < 64; k0 += 4) {   // hr @ Whh[0:64]
        v2f a, b;
        a.x = rs[nn][k0 + kg];
        a.y = rs[nn][k0 + kg + 1];
        b.x = Whh[(k0 + kg) * HID + col];
        b.y = Whh[(k0 + kg + 1) * HID + col];
        acc = wmma_f32_k4(a, b, acc);
    }
#pragma unroll
    for (int k0 = 0; k0 < 64; k0 += 4) {   // Thr @ Whh[64:128]
        v2f a, b;
        a.x = ts[nn][k0 + kg];
        a.y = ts[nn][k0 + kg + 1];
        b.x = Whh[(64 + k0 + kg) * HID + col];
        b.y = Whh[(64 + k0 + kg + 1) * HID + col];
        acc = wmma_f32_k4(a, b, acc);
    }

#pragma unroll
    for (int v = 0; v < 8; ++v) {
        int row = row0 + v + mh;
        float ht = tanhf(acc[v]);
        float z = Z[(size_t)row * HID + col];
        float hp = h[(size_t)row * HID + col];
        h[(size_t)row * HID + col] = z * hp + (1.0f - z) * ht;
    }
}

// u = h @ head_W + head_b  -> out[t]  (K=64, 3 outputs)
__global__ void head_kernel(const float* __restrict__ h, const float* __restrict__ W,
                            const float* __restrict__ b, float* __restrict__ out, int N) {
    int idx = blockIdx.x * blockDim.x + threadIdx.x;
    if (idx >= N * OUTF) return;
    int n = idx / OUTF, o = idx - n * OUTF;
    float acc = b[o];
#pragma unroll
    for (int c = 0; c < HID; ++c) acc = fmaf(h[n * HID + c], W[c * OUTF + o], acc);
    out[idx] = acc;
}

// ---------------- host launch ----------------

extern "C" void kernel_launch(void* const* d_in, const int* in_sizes, int n_in,
                              void* d_out, int out_size, void* d_ws, size_t ws_size,
                              hipStream_t stream) {
    const float* X_seq  = (const float*)d_in[0];
    const int*   edge   = (const int*)d_in[1];
    const float* Wx     = (const float*)d_in[2];
    const float* bx     = (const float*)d_in[3];
    const float* Wh     = (const float*)d_in[4];
    const float* bh     = (const float*)d_in[5];
    const float* head_W = (const float*)d_in[6];
    const float* head_b = (const float*)d_in[7];
    float* out = (float*)d_out;

    const int* src = edge;
    const int* dst = edge + EE;

    // Carve workspace
    char* base = (char*)d_ws;
    size_t off = 0;
    auto alloc = [&](size_t bytes) -> float* {
        float* p = (float*)(base + off);
        off = (off + bytes + 255) & ~(size_t)255;
        return p;
    };
    float* deg   = alloc((size_t)NN * 4);
    float* dis   = alloc((size_t)NN * 4);
    float* ew    = alloc((size_t)EE * 4);
    float* Wzr   = alloc(128 * 128 * 4);
    float* Whh   = alloc(128 * 64 * 4);
    float* Wxall = alloc(22 * 192 * 4);
    float* x_in  = alloc((size_t)NN * INF * 4);
    float* Tx    = alloc((size_t)NN * INF * 4);
    float* Gx    = alloc((size_t)NN * 192 * 4);
    float* h     = alloc((size_t)NN * HID * 4);
    float* Th    = alloc((size_t)NN * HID * 4);
    float* Zb    = alloc((size_t)NN * HID * 4);
    float* hrb   = alloc((size_t)NN * HID * 4);
    float* Thr   = alloc((size_t)NN * HID * 4);
    (void)ws_size;

    const int B = 256;
    auto nb = [](long long n, int b) { return (int)((n + b - 1) / b); };

    // Time-invariant graph prep
    zero_kernel<<<nb(NN, B), B, 0, stream>>>(deg, NN);
    deg_kernel<<<nb(EE, B), B, 0, stream>>>(src, deg, EE);
    dis_kernel<<<nb(NN, B), B, 0, stream>>>(deg, dis, NN);
    ew_kernel<<<nb(EE, B), B, 0, stream>>>(src, dst, dis, ew, EE);
    prep_w_kernel<<<nb(128 * 128 + 128 * 64 + 22 * 192, B), B, 0, stream>>>(Wx, Wh, Wzr, Whh, Wxall);

    // h0 = 0
    zero_kernel<<<nb((long long)NN * HID, B), B, 0, stream>>>(h, NN * HID);

    for (int t = 0; t < T_STEPS; ++t) {
        const float* u_prev = (t > 0) ? (out + (size_t)(t - 1) * NN * OUTF) : out;

        build_x_kernel<<<nb(NN, B), B, 0, stream>>>(X_seq, u_prev, x_in, t, NN);

        // Lhat * x_in
        zero_kernel<<<nb((long long)NN * INF, B), B, 0, stream>>>(Tx, NN * INF);
        spmm_kernel<<<nb((long long)EE * INF, B), B, 0, stream>>>(x_in, src, dst, ew, Tx, EE, INF);

        // x-side gate pre-activations (all 3 gates, incl. bx)
        gx_kernel<<<nb((long long)NN * 192, B), B, 0, stream>>>(x_in, Tx, Wxall, bx, Gx, NN);

        // Lhat * h
        zero_kernel<<<nb((long long)NN * HID, B), B, 0, stream>>>(Th, NN * HID);
        spmm_kernel<<<nb((long long)EE * HID, B), B, 0, stream>>>(h, src, dst, ew, Th, EE, HID);

        // Z, R gates (WMMA) + hr = h*R
        zr_kernel<<<NN / 16, 256, 0, stream>>>(h, Th, Wzr, Gx, bh, Zb, hrb, NN);

        // Lhat * (h*R)
        zero_kernel<<<nb((long long)NN * HID, B), B, 0, stream>>>(Thr, NN * HID);
        spmm_kernel<<<nb((long long)EE * HID, B), B, 0, stream>>>(hrb, src, dst, ew, Thr, EE, HID);

        // Candidate + state update (WMMA), in-place h
        hnew_kernel<<<NN / 16, 128, 0, stream>>>(hrb, Thr, Whh, Gx, bh + 128, Zb, h, NN);

        // Readout
        head_kernel<<<nb((long long)NN * OUTF, B), B, 0, stream>>>(
            h, head_W, head_b, out + (size_t)t * NN * OUTF, NN);
    }
}